// NeighborList_51032801411123
// MI455X (gfx1250) — compile-verified
//
#include <hip/hip_runtime.h>

// ---------------- problem constants (from reference) ----------------
#define N_ATOMS   20000
#define M_NBRS    32
#define N_CELLS   1000          // 10^3 cells, corner spacing 2.0
#define N_NBR     27
#define N_CAND    (N_NBR * M_NBRS)   // 864 candidates per atom
#define BIGF      1.0e10f
#define FLTMAXF   3.402823466e38f

#define CHUNK     2000          // atoms staged per TDM transfer (24 KB), 10 chunks
#define N_CHUNKS  (N_ATOMS / CHUNK)
#define TILES_PER_CHUNK (CHUNK / 16)    // 125

typedef __attribute__((ext_vector_type(2)))  float        v2f;
typedef __attribute__((ext_vector_type(8)))  float        v8f;
typedef __attribute__((ext_vector_type(4)))  unsigned int u32x4;
typedef __attribute__((ext_vector_type(4)))  int          i32x4;
typedef __attribute__((ext_vector_type(8)))  int          i32x8;

#if __has_builtin(__builtin_amdgcn_tensor_load_to_lds) && \
    __has_builtin(__builtin_amdgcn_s_wait_tensorcnt)
#define HAVE_TDM 1
#endif

// ---------------------------------------------------------------
// TDM: stage `nfloats` contiguous f32 from global into LDS.
// D# per ISA ch.8: group0 = {count=1, lds_addr, global_addr[56:0], type=2};
// group1 = 1-D tensor: data_size=4B, tensor_dim0=60000, tile_dim0=nfloats,
// tile_dim1=1, stride0=60000.  Groups 2/3 zero (<=2D tensor).
// ---------------------------------------------------------------
#ifdef HAVE_TDM
__device__ __forceinline__ void tdm_load_f32(const float* gsrc, unsigned lds_off,
                                             int nfloats) {
    unsigned long long ga = (unsigned long long)(size_t)gsrc;
    u32x4 g0;
    g0.x = 1u;                                            // count=1 (valid user D#)
    g0.y = lds_off;                                       // lds_addr (bytes)
    g0.z = (unsigned)(ga & 0xffffffffull);                // global_addr[31:0]
    g0.w = (unsigned)((ga >> 32) & 0x01ffffffull)         // global_addr[56:32]
         | (2u << 30);                                    // type=2 ("image")
    i32x8 g1;
    g1[0] = (int)(2u << 16);                              // data_size=2 -> 4 bytes
    g1[1] = (int)((unsigned)(N_ATOMS * 3) << 16);         // tensor_dim0[15:0]
    g1[2] = (int)(1u << 16);                              // tensor_dim1 = 1
    g1[3] = (int)((unsigned)nfloats << 16);               // tile_dim0
    g1[4] = 1;                                            // tile_dim1 = 1, tile_dim2 = 0
    g1[5] = N_ATOMS * 3;                                  // tensor_dim0_stride[31:0]
    g1[6] = 0;
    g1[7] = 0;
    i32x4 z4 = {0, 0, 0, 0};
#if defined(__clang_major__) && __clang_major__ >= 23
    i32x8 z8 = {0, 0, 0, 0, 0, 0, 0, 0};
    __builtin_amdgcn_tensor_load_to_lds(g0, g1, z4, z4, z8, 0);
#else
    __builtin_amdgcn_tensor_load_to_lds(g0, g1, z4, z4, 0);
#endif
}
#endif

// ---------------------------------------------------------------
// K0: cell corner coordinates.  meshgrid('xy') -> transpose -> reshape
// gives cells[n] = (2*i, 2*j, 2*k) with n = k*100 + i*10 + j.
// ---------------------------------------------------------------
__global__ void cells_kernel(float* __restrict__ cells) {
    int n = blockIdx.x * blockDim.x + threadIdx.x;
    if (n >= N_CELLS) return;
    int i = (n / 10) % 10;
    int j = n % 10;
    int k = n / 100;
    cells[n * 3 + 0] = 2.0f * (float)i;
    cells[n * 3 + 1] = 2.0f * (float)j;
    cells[n * 3 + 2] = 2.0f * (float)k;
}

// ---------------------------------------------------------------
// K1: per-cell 32 closest atoms.  One wave / 16 cells.
// Pipeline: TDM stages 2000-atom coord chunks into double-buffered LDS
// (TENSORcnt overlap), then V_WMMA_F32_16X16X4_F32 produces 16x16 dot
// tiles; dist^2 = |c|^2 + |a|^2 - 2 dot -> stable insertion into per-cell
// ascending 32-lists (== lax.top_k order incl. tie-breaks).
// ---------------------------------------------------------------
__global__ __launch_bounds__(32)
void closest_atoms_kernel(const float* __restrict__ coords,
                          const float* __restrict__ cells,
                          int* __restrict__ closest) {
    __shared__ float satoms[2][CHUNK * 3];   // 2 x 24 KB staging buffers
    __shared__ float cellc[16][4];           // 16 cells, padded K=4
    __shared__ float cc_s[16];               // |cell|^2
    __shared__ float distT[16 * 16];         // one WMMA tile of squared dists
    __shared__ float ld[16][M_NBRS];         // per-cell best distances (ascending)
    __shared__ int   li[16][M_NBRS];         // per-cell best atom indices

    const int lane     = threadIdx.x;        // wave32
    const int cellBase = blockIdx.x * 16;

    if (lane < 16) {
        int c = cellBase + lane;
        if (c >= N_CELLS) c = N_CELLS - 1;   // pad lanes: EXEC must stay full for WMMA
        float x = cells[c * 3 + 0];
        float y = cells[c * 3 + 1];
        float z = cells[c * 3 + 2];
        cellc[lane][0] = x; cellc[lane][1] = y;
        cellc[lane][2] = z; cellc[lane][3] = 0.0f;
        cc_s[lane] = x * x + y * y + z * z;
        for (int s = 0; s < M_NBRS; ++s) { ld[lane][s] = BIGF; li[lane][s] = 0; }
    }
    __syncthreads();

    // A fragment (constant over the whole loop):
    // lanes 0-15: M=lane, VGPR0=K0, VGPR1=K1 ; lanes 16-31: M=lane-16, VGPR0=K2, VGPR1=K3(=0)
    const int m  = lane & 15;
    const int kh = lane >> 4;
    v2f afrag;
    afrag.x = cellc[m][2 * kh + 0];
    afrag.y = cellc[m][2 * kh + 1];

    const int n = lane & 15;                 // atom column within tile

#ifdef HAVE_TDM
    const unsigned ldsOff[2] = {
        (unsigned)(size_t)(void*)&satoms[0][0],   // aperture-truncated = LDS byte offset
        (unsigned)(size_t)(void*)&satoms[1][0]
    };
    tdm_load_f32(coords, ldsOff[0], CHUNK * 3);   // prime chunk 0
#endif

    for (int c = 0; c < N_CHUNKS; ++c) {
        const int buf = c & 1;
#ifdef HAVE_TDM
        if (c + 1 < N_CHUNKS) {
            // issue next chunk's DMA, then wait for the current one (<=1 in flight)
            tdm_load_f32(coords + (size_t)(c + 1) * CHUNK * 3, ldsOff[buf ^ 1],
                         CHUNK * 3);
            __builtin_amdgcn_s_wait_tensorcnt(1);
        } else {
            __builtin_amdgcn_s_wait_tensorcnt(0);
        }
#else
        for (int i = lane; i < CHUNK * 3; i += 32)
            satoms[buf][i] = coords[(size_t)c * CHUNK * 3 + i];
#endif
        __syncthreads();

        for (int t = 0; t < TILES_PER_CHUNK; ++t) {
            const int la = t * 16 + n;                 // atom within chunk
            const float ax = satoms[buf][la * 3 + 0];
            const float ay = satoms[buf][la * 3 + 1];
            const float az = satoms[buf][la * 3 + 2];
            const float aa = ax * ax + ay * ay + az * az;

            // B fragment (4x16): lanes 0-15 carry K={0,1}, lanes 16-31 K={2,3(pad)}
            v2f bfrag;
            bfrag.x = (kh == 0) ? ax : az;
            bfrag.y = (kh == 0) ? ay : 0.0f;

            v8f acc = {0.f, 0.f, 0.f, 0.f, 0.f, 0.f, 0.f, 0.f};
            acc = __builtin_amdgcn_wmma_f32_16x16x4_f32(
                false, afrag, false, bfrag, (short)0, acc, false, false);

            // D layout: VGPR v -> M = v + 8*(lane>=16), N = lane&15
#pragma unroll
            for (int v = 0; v < 8; ++v) {
                const int mm = v + 8 * kh;
                distT[mm * 16 + n] = cc_s[mm] + aa - 2.0f * acc[v];
            }
            __syncthreads();

            // lanes 0-15 each maintain their cell's sorted 32-list (stable ->
            // same tie-break as lax.top_k: lower atom index first on ties)
            if (lane < 16) {
                for (int q = 0; q < 16; ++q) {
                    const float d = distT[lane * 16 + q];
                    if (d < ld[lane][M_NBRS - 1]) {
                        int p = M_NBRS - 1;
                        while (p > 0 && ld[lane][p - 1] > d) {
                            ld[lane][p] = ld[lane][p - 1];
                            li[lane][p] = li[lane][p - 1];
                            --p;
                        }
                        ld[lane][p] = d;
                        li[lane][p] = c * CHUNK + t * 16 + q;
                    }
                }
            }
            __syncthreads();
        }
    }

    if (lane < 16) {
        const int c = cellBase + lane;
        if (c < N_CELLS)
            for (int s = 0; s < M_NBRS; ++s)
                closest[c * M_NBRS + s] = li[lane][s];
    }
}

// ---------------------------------------------------------------
// K2: per-atom nearest cell, same WMMA tiling transposed:
// A = 16 atoms x K4, B = K4 x 16 cells, 63 cell tiles (clamp-padded so
// EXEC stays full; padded columns skipped in the argmin).  Strict '<'
// keeps the first minimum -> same tie-break as jnp.argmin.
// ---------------------------------------------------------------
__global__ __launch_bounds__(32)
void cells_for_atoms_kernel(const float* __restrict__ coords,
                            const float* __restrict__ cells,
                            int* __restrict__ cfa) {
    __shared__ float aa_s[16];
    __shared__ float distT[16 * 16];

    const int lane = threadIdx.x;
    const int m    = lane & 15;
    const int kh   = lane >> 4;
    const int n    = lane & 15;

    const int atom = blockIdx.x * 16 + m;        // 20000 = 16*1250, no remainder
    const float ax = coords[atom * 3 + 0];
    const float ay = coords[atom * 3 + 1];
    const float az = coords[atom * 3 + 2];
    if (lane < 16) aa_s[lane] = ax * ax + ay * ay + az * az;
    __syncthreads();

    v2f afrag;
    afrag.x = (kh == 0) ? ax : az;
    afrag.y = (kh == 0) ? ay : 0.0f;

    float bestd = FLTMAXF;                        // live only in lanes 0-15
    int   besti = 0;

    const int nTiles = (N_CELLS + 15) / 16;       // 63
    for (int t = 0; t < nTiles; ++t) {
        if (t + 1 < nTiles)
            __builtin_prefetch(&cells[(size_t)(t + 1) * 16 * 3], 0, 1);

        int cidx = t * 16 + n;
        int cl   = cidx < N_CELLS ? cidx : N_CELLS - 1;
        const float cx = cells[cl * 3 + 0];
        const float cy = cells[cl * 3 + 1];
        const float cz = cells[cl * 3 + 2];
        const float cc = cx * cx + cy * cy + cz * cz;

        v2f bfrag;
        bfrag.x = (kh == 0) ? cx : cz;
        bfrag.y = (kh == 0) ? cy : 0.0f;

        v8f acc = {0.f, 0.f, 0.f, 0.f, 0.f, 0.f, 0.f, 0.f};
        acc = __builtin_amdgcn_wmma_f32_16x16x4_f32(
            false, afrag, false, bfrag, (short)0, acc, false, false);

#pragma unroll
        for (int v = 0; v < 8; ++v) {
            const int mm = v + 8 * kh;
            distT[mm * 16 + n] = aa_s[mm] + cc - 2.0f * acc[v];
        }
        __syncthreads();

        if (lane < 16) {
            for (int q = 0; q < 16; ++q) {
                const int ci = t * 16 + q;
                if (ci < N_CELLS) {
                    const float d = distT[lane * 16 + q];
                    if (d < bestd) { bestd = d; besti = ci; }
                }
            }
        }
        __syncthreads();
    }

    if (lane < 16) cfa[blockIdx.x * 16 + lane] = besti;
}

// ---------------------------------------------------------------
// K3: per-cell 27 nearest cells (stable insertion -> top_k tie-break).
// ---------------------------------------------------------------
__global__ __launch_bounds__(256)
void neighbor_cells_kernel(const float* __restrict__ cells,
                           int* __restrict__ nbrc) {
    __shared__ float cs[N_CELLS * 3];
    for (int i = threadIdx.x; i < N_CELLS * 3; i += blockDim.x) cs[i] = cells[i];
    __syncthreads();

    const int c = blockIdx.x * blockDim.x + threadIdx.x;
    if (c >= N_CELLS) return;
    const float x = cs[c * 3 + 0];
    const float y = cs[c * 3 + 1];
    const float z = cs[c * 3 + 2];

    float bd[N_NBR];
    int   bi[N_NBR];
    for (int s = 0; s < N_NBR; ++s) { bd[s] = FLTMAXF; bi[s] = 0; }

    for (int o = 0; o < N_CELLS; ++o) {
        const float dx = cs[o * 3 + 0] - x;
        const float dy = cs[o * 3 + 1] - y;
        const float dz = cs[o * 3 + 2] - z;
        const float d = dx * dx + dy * dy + dz * dz;
        if (d < bd[N_NBR - 1]) {
            int p = N_NBR - 1;
            while (p > 0 && bd[p - 1] > d) {
                bd[p] = bd[p - 1]; bi[p] = bi[p - 1]; --p;
            }
            bd[p] = d; bi[p] = o;
        }
    }
    for (int s = 0; s < N_NBR; ++s) nbrc[c * N_NBR + s] = bi[s];
}

// ---------------------------------------------------------------
// K4: one block per atom.  Gather 864 candidates, first-occurrence dedup
// plus drop of row[0], masked distances, then 32 rounds of parallel argmin
// with (dist, position) tie-break == lax.top_k on -dists.
// Never materializes the 69 MB atoms_in_nbrs tensor.
// ---------------------------------------------------------------
__global__ __launch_bounds__(256)
void nbr_list_kernel(const float* __restrict__ coords,
                     const int* __restrict__ closest,
                     const int* __restrict__ cfa,
                     const int* __restrict__ nbrc,
                     int* __restrict__ out) {
    __shared__ int   cand[N_CAND];
    __shared__ float dist[N_CAND];
    __shared__ float rd[256];
    __shared__ int   rp[256];

    const int a = blockIdx.x;
    const int t = threadIdx.x;
    const int c = cfa[a];

    for (int j = t; j < N_CAND; j += 256) {
        const int nc = nbrc[c * N_NBR + j / M_NBRS];
        cand[j] = closest[nc * M_NBRS + (j % M_NBRS)];
    }
    __syncthreads();

    const int first = cand[0];
    const float x = coords[a * 3 + 0];
    const float y = coords[a * 3 + 1];
    const float z = coords[a * 3 + 2];

    for (int j = t; j < N_CAND; j += 256) {
        const int cj = cand[j];
        bool keep = (cj != first);
        if (keep) {
            for (int i = 0; i < j; ++i) {
                if (cand[i] == cj) { keep = false; break; }
            }
        }
        float d = BIGF;
        if (keep) {
            const float dx = coords[cj * 3 + 0] - x;
            const float dy = coords[cj * 3 + 1] - y;
            const float dz = coords[cj * 3 + 2] - z;
            d = dx * dx + dy * dy + dz * dz;
        }
        dist[j] = d;
    }
    __syncthreads();

    for (int r = 0; r < M_NBRS; ++r) {
        float bd = FLTMAXF;
        int   bp = N_CAND;
        for (int j = t; j < N_CAND; j += 256) {
            const float d = dist[j];
            if (d < bd || (d == bd && j < bp)) { bd = d; bp = j; }
        }
        rd[t] = bd; rp[t] = bp;
        __syncthreads();
        for (int s = 128; s > 0; s >>= 1) {
            if (t < s) {
                const float d2 = rd[t + s];
                const int   p2 = rp[t + s];
                if (d2 < rd[t] || (d2 == rd[t] && p2 < rp[t])) {
                    rd[t] = d2; rp[t] = p2;
                }
            }
            __syncthreads();
        }
        if (t == 0) {
            const int p = rp[0];
            out[a * M_NBRS + r] = cand[p];
            dist[p] = FLTMAXF;          // remove winner (BIGF < FLTMAXF keeps masked order)
        }
        __syncthreads();
    }
}

// ---------------------------------------------------------------
// Workspace layout (16B-aligned offsets), total ~328 KB:
//   [0)        cells           3000 f32  (12000 B, pad to 12032)
//   [12032)    closest_atoms  32000 i32 (128000 B)
//   [140032)   cells_for_atoms 20000 i32 (80000 B)
//   [220032)   neighbor_cells 27000 i32 (108000 B)
// ---------------------------------------------------------------
extern "C" void kernel_launch(void* const* d_in, const int* in_sizes, int n_in,
                              void* d_out, int out_size, void* d_ws, size_t ws_size,
                              hipStream_t stream) {
    const float* coords = (const float*)d_in[0];
    char* ws = (char*)d_ws;

    float* cells   = (float*)(ws);
    int*   closest = (int*)(ws + 12032);
    int*   cfa     = (int*)(ws + 140032);
    int*   nbrc    = (int*)(ws + 220032);
    int*   out     = (int*)d_out;   // reference output: int32 neighbor indices

    cells_kernel<<<(N_CELLS + 255) / 256, 256, 0, stream>>>(cells);
    closest_atoms_kernel<<<(N_CELLS + 15) / 16, 32, 0, stream>>>(coords, cells, closest);
    cells_for_atoms_kernel<<<N_ATOMS / 16, 32, 0, stream>>>(coords, cells, cfa);
    neighbor_cells_kernel<<<(N_CELLS + 255) / 256, 256, 0, stream>>>(cells, nbrc);
    nbr_list_kernel<<<N_ATOMS, 256, 0, stream>>>(coords, closest, cfa, nbrc, out);
}